// GALNLLLoss2d_29729763623124
// MI455X (gfx1250) — compile-verified
//
#include <hip/hip_runtime.h>
#include <math.h>

typedef __attribute__((ext_vector_type(2))) float v2f;
typedef __attribute__((ext_vector_type(8))) float v8f;

#define LOG2_F    0.6931471805599453f
#define LOG2PI_F  1.8378770664093453f

// ds_swizzle SWAPX16: offset[15]=0 (group-of-32), xor=0x10, or=0, and=0x1f
#define SWIZZLE_SWAPX16 0x401F

// Wave32 sum via WMMA ones-matrix trick:
// A (16x4): lane L<16 -> A[L][0]=v, lane L>=16 -> A[L-16][2]=v, other K = 0.
// B (4x16) = ones.  D[m][n] = v_m + v_{m+16}.
// D layout: VGPR r holds D[r][lane] (lanes 0-15) / D[r+8][lane-16] (16-31),
// so summing the 8 D VGPRs gives each half-wave's sum; one ds_swizzle
// (SWAPX16) + add completes the full 32-lane sum.
__device__ __forceinline__ float wave_sum32(float v) {
    v2f a; a[0] = v;     a[1] = 0.0f;
    v2f b; b[0] = 1.0f;  b[1] = 1.0f;
    v8f c = {};
    c = __builtin_amdgcn_wmma_f32_16x16x4_f32(
            /*neg_a=*/false, a, /*neg_b=*/false, b,
            /*c_mod=*/(short)0, c, /*reuse_a=*/false, /*reuse_b=*/false);
    float s = ((c[0] + c[1]) + (c[2] + c[3])) + ((c[4] + c[5]) + (c[6] + c[7]));
    int other = __builtin_amdgcn_ds_swizzle(__float_as_int(s), SWIZZLE_SWAPX16);
    s += __int_as_float(other);
    return s;                      // full 32-lane sum in every lane
}

// 256-thread block reduction; result valid in threadIdx.x == 0.
__device__ __forceinline__ float block_sum256(float v) {
    __shared__ float lds[8];
    float s = wave_sum32(v);
    int lane = threadIdx.x & 31;
    int wave = threadIdx.x >> 5;
    if (lane == 0) lds[wave] = s;
    __syncthreads();
    float t = 0.0f;
    if (threadIdx.x == 0) {
#pragma unroll
        for (int i = 0; i < 8; ++i) t += lds[i];
    }
    return t;
}

// -log_prob for one element.
__device__ __forceinline__ float neg_logprob(float a, float cL, float d,
                                             float m0, float m1,
                                             float dx0, float dx1) {
    float inva = 1.0f / a;
    float invd = 1.0f / d;

    // L^{-1} v by forward substitution: [v0/a, (v1 - c*(v0/a))/d]
    float pm = m0 * inva;
    float qm = (m1 - cL * pm) * invd;
    float pd = dx0 * inva;
    float qd = (dx1 - cL * pd) * invd;

    float mSm = fmaf(pm, pm, qm * qm);
    float xSx = fmaf(pd, pd, qd * qd);
    float xSm = fmaf(pd, pm, qd * qm);

    float tm = 2.0f + mSm;
    float z  = sqrtf(tm * xSx);
    float lz = logf(z);                          // shared by both branches

    // logdet(Sinv) = -2*log|L00*L11|
    float logdet = -2.0f * logf(fabsf(a * d));

    // ---- kve1 log-term:  log(exp(z)*K1(z)) - z ----
    bool small = (z <= 2.0f);

    // small branch: log(z*K1(z)) - log(z); exp(z) cancels exactly.
    float t  = z * 0.26666667f;  t = t * t;       // (z/3.75)^2
    float i1 = z * fmaf(t, fmaf(t, fmaf(t, fmaf(t, fmaf(t,
                    fmaf(t, 0.00032411f, 0.00301532f),
                    0.02658733f), 0.15084934f), 0.51498869f),
                    0.87890594f), 0.5f);
    float t2 = 0.25f * z * z;
    float p2 = fmaf(t2, fmaf(t2, fmaf(t2, fmaf(t2,
                    fmaf(t2, -0.00004686f, -0.00110404f),
                    -0.01919402f), -0.18156897f), -0.67278579f),
                    0.15443144f);
    float zk1 = fmaf(z * (lz - LOG2_F), i1, fmaf(t2, p2, 1.0f));

    // large branch: log(poly(2/z)) - 0.5*log(z) - z
    float u  = 2.0f / z;
    float p3 = fmaf(u, fmaf(u, fmaf(u, fmaf(u, fmaf(u,
                    fmaf(u, -0.00068245f, 0.00325614f),
                    -0.00780353f), 0.01504268f), -0.03655620f),
                    0.23498619f), 1.25331414f);

    float r     = small ? (zk1 / z) : p3;        // single shared log below
    float tail  = small ? 0.0f : fmaf(0.5f, lz, z);
    float kterm = logf(r) - tail;

    float lp = LOG2_F + xSm - LOG2PI_F
             + 0.5f * logdet
             + 0.5f * logf(xSx / tm)
             + kterm;
    return -lp;
}

__global__ __launch_bounds__(256)
void gal_nll_main(const float4* __restrict__ loc2, const float4* __restrict__ m2,
                  const float4* __restrict__ L,    const float4* __restrict__ x2,
                  float* __restrict__ partials, int Btot) {
    const int tid    = blockIdx.x * blockDim.x + threadIdx.x;
    const int stride = gridDim.x * blockDim.x;
    const int pairs  = Btot >> 1;          // two elements per iteration
    const int nFull  = pairs / stride;     // uniform trip count (no exec churn)

    float acc = 0.0f;

    for (int j = 0; j < nFull; ++j) {
        int p = tid + j * stride;
        float4 lo = loc2[p];               // elems 2p, 2p+1: all b128 loads
        float4 mm = m2[p];
        float4 xx = x2[p];
        float4 La = L[2 * p + 0];
        float4 Lb = L[2 * p + 1];

        acc += neg_logprob(La.x, La.z, La.w, mm.x, mm.y,
                           xx.x - lo.x, xx.y - lo.y);
        acc += neg_logprob(Lb.x, Lb.z, Lb.w, mm.z, mm.w,
                           xx.z - lo.z, xx.w - lo.w);
    }
    // tail (empty when pairs % stride == 0, e.g. B = 2^22)
    for (int p = nFull * stride + tid; p < pairs; p += stride) {
        float4 lo = loc2[p];
        float4 mm = m2[p];
        float4 xx = x2[p];
        float4 La = L[2 * p + 0];
        float4 Lb = L[2 * p + 1];
        acc += neg_logprob(La.x, La.z, La.w, mm.x, mm.y,
                           xx.x - lo.x, xx.y - lo.y);
        acc += neg_logprob(Lb.x, Lb.z, Lb.w, mm.z, mm.w,
                           xx.z - lo.z, xx.w - lo.w);
    }

    float bs = block_sum256(acc);
    if (threadIdx.x == 0) partials[blockIdx.x] = bs;
}

__global__ __launch_bounds__(256)
void gal_nll_finish(const float* __restrict__ partials, int n,
                    float* __restrict__ out, float invB) {
    float acc = 0.0f;
    for (int i = threadIdx.x; i < n; i += blockDim.x) acc += partials[i];
    float bs = block_sum256(acc);
    if (threadIdx.x == 0) out[0] = bs * invB;
}

extern "C" void kernel_launch(void* const* d_in, const int* in_sizes, int n_in,
                              void* d_out, int out_size, void* d_ws, size_t ws_size,
                              hipStream_t stream) {
    // setup_inputs order: loc (B,2), m (B,2), L (B,2,2), x (B,2) — all float32
    const float4* loc = (const float4*)d_in[0];
    const float4* m   = (const float4*)d_in[1];
    const float4* L   = (const float4*)d_in[2];
    const float4* x   = (const float4*)d_in[3];
    float* out        = (float*)d_out;
    float* partials   = (float*)d_ws;

    int Btot = in_sizes[0] / 2;                 // 4,194,304
    const int THREADS = 256;
    const int BLOCKS  = 2048;                   // 16K waves in flight; 8 KB partials

    gal_nll_main<<<BLOCKS, THREADS, 0, stream>>>(loc, m, L, x, partials, Btot);
    gal_nll_finish<<<1, THREADS, 0, stream>>>(partials, BLOCKS, out,
                                              1.0f / (float)Btot);
}